// HybridGNN_63032940036573
// MI455X (gfx1250) — compile-verified
//
#include <hip/hip_runtime.h>
#include <math.h>

// HybridGNN for MI455X (gfx1250): GCN -> ReLU -> GAT -> ELU -> GCN
// fp32 end-to-end; dense GEMMs via V_WMMA_F32_16X16X4_F32 (wave32 WMMA).

#define IN_C   128
#define HID    64
#define HEADS  4
#define HHC    256          // HEADS*HID
#define OUT_C  2
#define NEG_SLOPE 0.2f

typedef float v2f __attribute__((ext_vector_type(2)));
typedef float v8f __attribute__((ext_vector_type(8)));

// ---------------- utility kernels ----------------

__global__ void fill_kernel(float* __restrict__ p, int n, float v) {
  int i = blockIdx.x * blockDim.x + threadIdx.x;
  if (i < n) p[i] = v;
}

__global__ void init_out_kernel(float* __restrict__ out, const float* __restrict__ b3, int total) {
  int i = blockIdx.x * blockDim.x + threadIdx.x;
  if (i < total) out[i] = b3[i & (OUT_C - 1)];
}

// ---------------- WMMA GEMM: C[M,NC] = A[M,K] * B[K,NC] ----------------
// One wave per 16x16 output tile. M % 16 == 0, K % 4 == 0, NC % 16 == 0.
// f32 A-frag layout (16x4): lanes 0-15 hold K={k,k+1}, lanes 16-31 K={k+2,k+3}, M = lane%16.
// f32 B-frag layout (4x16): lanes 0-15 hold rows K={k,k+1}, lanes 16-31 K={k+2,k+3}, N = lane%16.
// C/D layout: VGPR r -> M = r + 8*(lane/16), N = lane%16.
__global__ void gemm_wmma_f32(const float* __restrict__ A, const float* __restrict__ B,
                              float* __restrict__ C, int M, int K, int NC) {
  const int lane = threadIdx.x & 31;
  const int wave = (blockIdx.x * blockDim.x + threadIdx.x) >> 5;
  const int tilesN = NC >> 4;
  const int totalTiles = (M >> 4) * tilesN;
  if (wave >= totalTiles) return;               // uniform per-wave exit: EXEC stays all-1s
  const int tm = wave / tilesN;
  const int tn = wave - tm * tilesN;
  const int half = lane >> 4;                   // 0 or 1
  const int l16  = lane & 15;
  const int rowA = tm * 16 + l16;
  const int colB = tn * 16 + l16;
  const float* __restrict__ Ap = A + (size_t)rowA * K;

  v8f acc = {};
  for (int k = 0; k < K; k += 4) {
    const int ka = k + (half << 1);
    v2f a, b;
    a.x = Ap[ka];
    a.y = Ap[ka + 1];
    b.x = B[(size_t)ka * NC + colB];
    b.y = B[(size_t)(ka + 1) * NC + colB];
    acc = __builtin_amdgcn_wmma_f32_16x16x4_f32(
        /*neg_a=*/false, a, /*neg_b=*/false, b,
        /*c_mod=*/(short)0, acc, /*reuse_a=*/false, /*reuse_b=*/false);
  }
#pragma unroll
  for (int r = 0; r < 8; ++r) {
    const int row = tm * 16 + (half << 3) + r;
    C[(size_t)row * NC + colB] = acc[r];
  }
}

// ---------------- graph helpers ----------------

__device__ __forceinline__ void edge_sd(const int* __restrict__ ei, int E, int e, int& s, int& d) {
  if (e < E) { s = ei[e]; d = ei[E + e]; }      // edge_index rows [2,E]
  else       { s = d = e - E; }                 // self loop
}

__device__ __forceinline__ void atomicMaxFloat(float* addr, float val) {
  // works with addr initialized to -inf
  if (val >= 0.0f) atomicMax((int*)addr, __float_as_int(val));
  else             atomicMin((unsigned int*)addr, __float_as_uint(val));
}

__global__ void deg_kernel(const int* __restrict__ ei, int E, int E2, float* __restrict__ deg) {
  int e = blockIdx.x * blockDim.x + threadIdx.x;
  if (e >= E2) return;
  int s, d; edge_sd(ei, E, e, s, d);
  atomicAdd(&deg[d], 1.0f);
}

__global__ void dinv_kernel(float* __restrict__ deg, int n) {
  int i = blockIdx.x * blockDim.x + threadIdx.x;
  if (i >= n) return;
  float v = deg[i];
  deg[i] = (v > 0.0f) ? rsqrtf(v) : 0.0f;       // in-place: deg -> dinv
}

// out[dst, :C] += xw[src, :C] * dinv[src]*dinv[dst]; 1 thread per (edge, 4-channel group)
__global__ void gcn_scatter(const float* __restrict__ xw, const float* __restrict__ dinv,
                            const int* __restrict__ ei, int E, int E2,
                            float* __restrict__ outacc, int C) {
  int tid = blockIdx.x * blockDim.x + threadIdx.x;
  const int groups = C >> 2;
  int e = tid / groups;
  if (e >= E2) return;
  int g = tid - e * groups;
  int s, d; edge_sd(ei, E, e, s, d);
  __builtin_prefetch(xw + (size_t)s * C, 0, 0); // global_prefetch_b8
  float w = dinv[s] * dinv[d];
  float4 v = *(const float4*)(xw + (size_t)s * C + g * 4);
  float* o = outacc + (size_t)d * C + g * 4;
  atomicAdd(o + 0, v.x * w);
  atomicAdd(o + 1, v.y * w);
  atomicAdd(o + 2, v.z * w);
  atomicAdd(o + 3, v.w * w);
}

__global__ void relu_bias_kernel(float* __restrict__ h, const float* __restrict__ b, int total) {
  int i = blockIdx.x * blockDim.x + threadIdx.x;
  if (i >= total) return;
  float v = h[i] + b[i & (HID - 1)];
  h[i] = v > 0.0f ? v : 0.0f;
}

// per-(node, head) attention logits: a_s = <xw2[i,h,:], att_src[h,:]>, same for a_d
__global__ void att_score_kernel(const float* __restrict__ xw2,
                                 const float* __restrict__ att_src,
                                 const float* __restrict__ att_dst,
                                 float* __restrict__ as_, float* __restrict__ ad_, int n) {
  int tid = blockIdx.x * blockDim.x + threadIdx.x;
  if (tid >= n * HEADS) return;
  int i = tid >> 2, hh = tid & 3;
  const float* v = xw2 + (size_t)i * HHC + hh * HID;
  const float* ws = att_src + hh * HID;
  const float* wd = att_dst + hh * HID;
  float ss = 0.0f, sd = 0.0f;
  for (int c = 0; c < HID; ++c) { float x = v[c]; ss += x * ws[c]; sd += x * wd[c]; }
  as_[tid] = ss; ad_[tid] = sd;
}

__device__ __forceinline__ float edge_logit(const float* as_, const float* ad_,
                                            int s, int d, int hh) {
  float el = as_[s * HEADS + hh] + ad_[d * HEADS + hh];
  return el >= 0.0f ? el : NEG_SLOPE * el;      // leaky_relu
}

__global__ void edge_max_kernel(const float* __restrict__ as_, const float* __restrict__ ad_,
                                const int* __restrict__ ei, int E, int E2,
                                float* __restrict__ m_) {
  int tid = blockIdx.x * blockDim.x + threadIdx.x;
  int e = tid >> 2;
  if (e >= E2) return;
  int hh = tid & 3;
  int s, d; edge_sd(ei, E, e, s, d);
  atomicMaxFloat(&m_[d * HEADS + hh], edge_logit(as_, ad_, s, d, hh));
}

__global__ void m_fix_kernel(float* __restrict__ m_, int total) {
  int i = blockIdx.x * blockDim.x + threadIdx.x;
  if (i >= total) return;
  float v = m_[i];
  if (!(fabsf(v) < INFINITY)) m_[i] = 0.0f;
}

__global__ void edge_den_kernel(const float* __restrict__ as_, const float* __restrict__ ad_,
                                const float* __restrict__ m_,
                                const int* __restrict__ ei, int E, int E2,
                                float* __restrict__ den) {
  int tid = blockIdx.x * blockDim.x + threadIdx.x;
  int e = tid >> 2;
  if (e >= E2) return;
  int hh = tid & 3;
  int s, d; edge_sd(ei, E, e, s, d);
  float ex = expf(edge_logit(as_, ad_, s, d, hh) - m_[d * HEADS + hh]);
  atomicAdd(&den[d * HEADS + hh], ex);
}

// gat[dst, :256] += xw2[src, :256] * alpha(edge, head); 1 thread per (edge, 4-channel group)
__global__ void gat_scatter_kernel(const float* __restrict__ xw2,
                                   const float* __restrict__ as_, const float* __restrict__ ad_,
                                   const float* __restrict__ m_, const float* __restrict__ den,
                                   const int* __restrict__ ei, int E, int E2,
                                   float* __restrict__ gat) {
  int tid = blockIdx.x * blockDim.x + threadIdx.x;
  int e = tid >> 6;                              // 64 quad-groups of 4 ch = 256 ch
  if (e >= E2) return;
  int g = tid & 63;
  int hh = g >> 4;                               // 16 quads per head
  int s, d; edge_sd(ei, E, e, s, d);
  float el = edge_logit(as_, ad_, s, d, hh);
  float alpha = expf(el - m_[d * HEADS + hh]) / (den[d * HEADS + hh] + 1e-16f);
  float4 v = *(const float4*)(xw2 + (size_t)s * HHC + g * 4);
  float* o = gat + (size_t)d * HHC + g * 4;
  atomicAdd(o + 0, v.x * alpha);
  atomicAdd(o + 1, v.y * alpha);
  atomicAdd(o + 2, v.z * alpha);
  atomicAdd(o + 3, v.w * alpha);
}

// t[i,c] = sum_k elu(gat[i,k] + b2[k]) * W3[k,c]   (NC=2: scalar VALU GEMM)
__global__ void gemm3_elu_kernel(const float* __restrict__ gat, const float* __restrict__ b2,
                                 const float* __restrict__ W3, float* __restrict__ t, int n) {
  int tid = blockIdx.x * blockDim.x + threadIdx.x;
  if (tid >= n * OUT_C) return;
  int i = tid >> 1, c = tid & 1;
  const float* gi = gat + (size_t)i * HHC;
  float acc = 0.0f;
  for (int k = 0; k < HHC; ++k) {
    float v = gi[k] + b2[k];
    v = v > 0.0f ? v : (expf(v) - 1.0f);         // elu
    acc += v * W3[k * OUT_C + c];
  }
  t[tid] = acc;
}

__global__ void final_scatter_kernel(const float* __restrict__ t, const float* __restrict__ dinv,
                                     const int* __restrict__ ei, int E, int E2,
                                     float* __restrict__ out) {
  int tid = blockIdx.x * blockDim.x + threadIdx.x;
  int e = tid >> 1;
  if (e >= E2) return;
  int c = tid & 1;
  int s, d; edge_sd(ei, E, e, s, d);
  float w = dinv[s] * dinv[d];
  atomicAdd(&out[d * OUT_C + c], t[s * OUT_C + c] * w);
}

// ---------------- launch ----------------

static inline unsigned gridFor(long long n) { return (unsigned)((n + 255) / 256); }

extern "C" void kernel_launch(void* const* d_in, const int* in_sizes, int n_in,
                              void* d_out, int out_size, void* d_ws, size_t ws_size,
                              hipStream_t stream) {
  const float* x       = (const float*)d_in[0];
  const int*   ei      = (const int*)d_in[1];     // [2,E] int32 (JAX x64 disabled)
  const float* W1      = (const float*)d_in[2];
  const float* b1      = (const float*)d_in[3];
  const float* W2      = (const float*)d_in[4];
  const float* att_src = (const float*)d_in[5];
  const float* att_dst = (const float*)d_in[6];
  const float* b2      = (const float*)d_in[7];
  const float* W3      = (const float*)d_in[8];
  const float* b3      = (const float*)d_in[9];
  float* out = (float*)d_out;

  const int N  = in_sizes[0] / IN_C;   // 100000 (multiple of 16)
  const int E  = in_sizes[1] / 2;      // 1.6M
  const int E2 = E + N;                // with self loops

  // workspace layout (floats)
  float* p    = (float*)d_ws;
  float* xw1  = p; p += (size_t)N * HID;
  float* h    = p; p += (size_t)N * HID;
  float* xw2  = p; p += (size_t)N * HHC;
  float* gat  = p; p += (size_t)N * HHC;
  float* dinv = p; p += N;                       // deg, then dinv in-place
  float* as_  = p; p += (size_t)N * HEADS;
  float* ad_  = p; p += (size_t)N * HEADS;
  float* m_   = p; p += (size_t)N * HEADS;
  float* den  = p; p += (size_t)N * HEADS;
  float* t    = p; p += (size_t)N * OUT_C;

  const dim3 B(256);

  // init accumulators
  fill_kernel<<<gridFor(N), B, 0, stream>>>(dinv, N, 0.0f);
  fill_kernel<<<gridFor((long long)N * HID), B, 0, stream>>>(h, N * HID, 0.0f);
  fill_kernel<<<gridFor((long long)N * HHC), B, 0, stream>>>(gat, N * HHC, 0.0f);
  fill_kernel<<<gridFor((long long)N * HEADS), B, 0, stream>>>(den, N * HEADS, 0.0f);
  fill_kernel<<<gridFor((long long)N * HEADS), B, 0, stream>>>(m_, N * HEADS, -INFINITY);
  init_out_kernel<<<gridFor((long long)N * OUT_C), B, 0, stream>>>(out, b3, N * OUT_C);

  // degrees -> dinv
  deg_kernel<<<gridFor(E2), B, 0, stream>>>(ei, E, E2, dinv);
  dinv_kernel<<<gridFor(N), B, 0, stream>>>(dinv, N);

  // ---- layer 1: GCN(128->64) + ReLU ----
  {
    long long waves = ((long long)(N >> 4)) * (HID >> 4);
    gemm_wmma_f32<<<gridFor(waves * 32), B, 0, stream>>>(x, W1, xw1, N, IN_C, HID);
  }
  gcn_scatter<<<gridFor((long long)E2 * (HID / 4)), B, 0, stream>>>(xw1, dinv, ei, E, E2, h, HID);
  relu_bias_kernel<<<gridFor((long long)N * HID), B, 0, stream>>>(h, b1, N * HID);

  // ---- layer 2: GAT(64 -> 4x64) + ELU ----
  {
    long long waves = ((long long)(N >> 4)) * (HHC >> 4);
    gemm_wmma_f32<<<gridFor(waves * 32), B, 0, stream>>>(h, W2, xw2, N, HID, HHC);
  }
  att_score_kernel<<<gridFor((long long)N * HEADS), B, 0, stream>>>(xw2, att_src, att_dst, as_, ad_, N);
  edge_max_kernel<<<gridFor((long long)E2 * HEADS), B, 0, stream>>>(as_, ad_, ei, E, E2, m_);
  m_fix_kernel<<<gridFor((long long)N * HEADS), B, 0, stream>>>(m_, N * HEADS);
  edge_den_kernel<<<gridFor((long long)E2 * HEADS), B, 0, stream>>>(as_, ad_, m_, ei, E, E2, den);
  gat_scatter_kernel<<<gridFor((long long)E2 * (HHC / 4)), B, 0, stream>>>(xw2, as_, ad_, m_, den,
                                                                           ei, E, E2, gat);

  // ---- layer 3: GCN(256->2), elu fused into the small GEMM ----
  gemm3_elu_kernel<<<gridFor((long long)N * OUT_C), B, 0, stream>>>(gat, b2, W3, t, N);
  final_scatter_kernel<<<gridFor((long long)E2 * OUT_C), B, 0, stream>>>(t, dinv, ei, E, E2, out);
}